// UpSamplingUsingArgIndices_43980465111285
// MI455X (gfx1250) — compile-verified
//
#include <hip/hip_runtime.h>

// -----------------------------------------------------------------------------
// Max-unpooling scatter-add for x:(16,128,128,64) f32, idx:(same shape) int64
// out:(16,256,256,64) f32.
//   out_flat = ((i >> 20) << 22) + (idx & 0x3FFFFF)   then atomic accumulate.
// Memory-bound op: no WMMA applies (0 FLOPs of matrix work). CDNA5 paths used:
//   * TDM tensor_store_from_lds (iterated DMA memset of the 256MB output)
//   * non-temporal (th:NT) 128-bit streaming loads
//   * s_wait_dscnt / s_wait_tensorcnt split counters
// -----------------------------------------------------------------------------

typedef __attribute__((ext_vector_type(4))) float        v4f;
typedef __attribute__((ext_vector_type(4))) int          v4i;
typedef __attribute__((ext_vector_type(4))) unsigned int v4u;
typedef __attribute__((ext_vector_type(8))) int          v8i;

#define IMG_LOG2   20u                 // H*W*C = 2^20 elements per batch (input)
#define OUTB_LOG2  22u                 // 2H*2W*C = 2^22 elements per batch (output)
#define N_ELEMS    (1u << 24)          // 16 * 2^20 input elements
#define OUT_ELEMS  (1u << 26)          // 16 * 2^22 output elements

// ---- TDM zero-fill parameters ----
#define ZTILE_QW   2048u                               // 16 KB of u64 zeros in LDS
#define ZITER      64u                                 // iterations per TDM op
#define ZCHUNK_B   (ZTILE_QW * 8u * ZITER)             // 1 MB per block
#define ZBLOCKS    ((OUT_ELEMS * 4u) / ZCHUNK_B)       // 256 blocks cover 256 MB

// One wave per block. Zero a 16KB LDS tile, then issue a single iterated
// TENSOR_STORE_FROM_LDS that replays the tile 64x into global memory
// (lds_addr_increment = 0, global_addr_increment = tile size).
__global__ __launch_bounds__(32) void tdm_zero_kernel(float* __restrict__ out) {
#if defined(__HIP_DEVICE_COMPILE__) && __has_builtin(__builtin_amdgcn_tensor_store_from_lds)
    __shared__ unsigned long long zbuf[ZTILE_QW];
    const unsigned lane = threadIdx.x;
    for (unsigned j = lane; j < ZTILE_QW; j += 32u) zbuf[j] = 0ull;
    // make the LDS zeros visible to the TDM engine before it reads them
    asm volatile("s_wait_dscnt 0x0" ::: "memory");

    const unsigned long long ga =
        (unsigned long long)(__SIZE_TYPE__)out + (unsigned long long)blockIdx.x * ZCHUNK_B;
    const unsigned lds_off = (unsigned)(__SIZE_TYPE__)&zbuf[0];   // low 32 bits = LDS offset

    // ---- D# group 0: flags | lds_addr | global_addr | type=2 ----
    v4u g0;
    g0.x = 0x1u;                                       // count=1 (valid), user mode
    g0.y = lds_off;                                    // lds_addr (bytes)
    g0.z = (unsigned)(ga & 0xFFFFFFFFull);             // global_addr[31:0]
    g0.w = (unsigned)((ga >> 32) & 0x1FFFFFFull)       // global_addr[56:32]
         | (2u << 30);                                 // type = 2 ("image")

    // ---- D# group 1: data_size=8B, iterate_enable, dims/tile/stride ----
    v8i g1;
    g1[0] = (int)((3u << 16) | (1u << 19));            // data_size=3 (8B), iterate_enable=1
    g1[1] = (int)0xFFFF0000u;                          // tensor_dim0[15:0]=0xFFFF (dim0=0x7FFFFFFF)
    g1[2] = (int)0xFFFF7FFFu;                          // tensor_dim0[31:16]=0x7FFF, tensor_dim1[15:0]=0xFFFF
    g1[3] = (int)((ZTILE_QW << 16) | 0x7FFFu);         // tensor_dim1[31:16]=0x7FFF, tile_dim0=2048
    g1[4] = 1;                                         // tile_dim1=1, tile_dim2=0
    g1[5] = (int)ZTILE_QW;                             // tensor_dim0_stride[31:0]
    g1[6] = 0;                                         // stride0[47:32]=0, stride1[15:0]=0
    g1[7] = 0;

    // ---- D# group 2 (iterate mode): lds_inc=0, global_inc=tile, count=ZITER ----
    v4i g2;
    g2[0] = 1;                                         // tensor_dim2 (unused, nonzero)
    g2[1] = 0;                                         // lds_addr_increment = 0 (replay zero tile)
    g2[2] = (int)ZTILE_QW;                             // global_addr_increment (data_size units)
    g2[3] = (int)((ZITER - 1u) << 16);                 // iterate_count = 64 iterations

    v4i g3 = {0, 0, 0, 0};                             // group 3 unused (no dim3/dim4)
    v8i g4 = {0, 0, 0, 0, 0, 0, 0, 0};                 // extra operand (clang-23 6-arg
                                                       // form); ISA VADDR4 is unused ->
                                                       // zero-filled

    __builtin_amdgcn_tensor_store_from_lds(g0, g1, g2, g3, g4, /*cpol=*/0);
    __builtin_amdgcn_s_wait_tensorcnt(0);
#else
    // Fallback: plain 128-bit stores over this block's 1 MB chunk.
    v4f* o = (v4f*)out + (__SIZE_TYPE__)blockIdx.x * (ZCHUNK_B / 16u);
    const v4f z = {0.f, 0.f, 0.f, 0.f};
    for (unsigned j = threadIdx.x; j < ZCHUNK_B / 16u; j += 32u) o[j] = z;
#endif
}

// 8 elements per thread: 2x B128 NT loads of x, 4x B128 NT loads of int64
// indices (low dword of each int64 is the flat argmax), 8 f32 atomics into the
// per-batch 16 MB output window (L2-resident under linear work order).
__global__ __launch_bounds__(256) void scatter_add_kernel(const v4f* __restrict__ xv,
                                                          const v4i* __restrict__ iv,
                                                          float* __restrict__ out) {
    const unsigned t  = blockIdx.x * 256u + threadIdx.x;   // 0 .. 2M-1
    const unsigned e0 = t << 3;                            // first of 8 elements

    const v4f x0 = __builtin_nontemporal_load(xv + (__SIZE_TYPE__)t * 2u + 0u);
    const v4f x1 = __builtin_nontemporal_load(xv + (__SIZE_TYPE__)t * 2u + 1u);
    const v4i q0 = __builtin_nontemporal_load(iv + (__SIZE_TYPE__)t * 4u + 0u);
    const v4i q1 = __builtin_nontemporal_load(iv + (__SIZE_TYPE__)t * 4u + 1u);
    const v4i q2 = __builtin_nontemporal_load(iv + (__SIZE_TYPE__)t * 4u + 2u);
    const v4i q3 = __builtin_nontemporal_load(iv + (__SIZE_TYPE__)t * 4u + 3u);

    // all 8 elements share one batch (2^20 is a multiple of 8)
    float* ob = out + ((__SIZE_TYPE__)(e0 >> IMG_LOG2) << OUTB_LOG2);
    const unsigned m = (1u << OUTB_LOG2) - 1u;             // idx mod 2^22

    atomicAdd(ob + ((unsigned)q0.x & m), x0.x);
    atomicAdd(ob + ((unsigned)q0.z & m), x0.y);
    atomicAdd(ob + ((unsigned)q1.x & m), x0.z);
    atomicAdd(ob + ((unsigned)q1.z & m), x0.w);
    atomicAdd(ob + ((unsigned)q2.x & m), x1.x);
    atomicAdd(ob + ((unsigned)q2.z & m), x1.y);
    atomicAdd(ob + ((unsigned)q3.x & m), x1.z);
    atomicAdd(ob + ((unsigned)q3.z & m), x1.w);
}

extern "C" void kernel_launch(void* const* d_in, const int* in_sizes, int n_in,
                              void* d_out, int out_size, void* d_ws, size_t ws_size,
                              hipStream_t stream) {
    (void)in_sizes; (void)n_in; (void)out_size; (void)d_ws; (void)ws_size;

    const v4f* xv  = (const v4f*)d_in[0];   // 16M f32, read as float4
    const v4i* iv  = (const v4i*)d_in[1];   // 16M int64, read as dword4 (2 int64 each)
    float*     out = (float*)d_out;         // 64M f32

    // Pass 1: DMA memset of the 256 MB output via TDM (harness poisons d_out;
    // we must re-zero every call — deterministic, graph-capture safe).
    tdm_zero_kernel<<<ZBLOCKS, 32, 0, stream>>>(out);

    // Pass 2: streaming scatter-add. 16M elems / 8 per thread / 256 per block.
    scatter_add_kernel<<<(N_ELEMS / 8u) / 256u, 256, 0, stream>>>(xv, iv, out);
}